// TCR_45552423141922
// MI455X (gfx1250) — compile-verified
//
#include <hip/hip_runtime.h>
#include <math.h>

// ---------------------------------------------------------------------------
// Affine warp (kornia warp_affine, bilinear, zeros padding) for [B,3,H,W] f32.
// Memory-bound: ~200 MB traffic -> ~8.6 us floor at 23.3 TB/s on MI455X.
// WMMA is inapplicable (data-dependent gather, 0.0075 FLOP/byte); optimize the
// data-movement path instead: scalar-uniform param loads, NT output stores,
// fully-unrolled gather with max MLP, speculative prefetch of next planes.
// Trig uses hardware v_sin/v_cos (args bounded to |x| <= 0.7 rad), avoiding
// the ~1000-instruction libm Payne-Hanek reduction.
// ---------------------------------------------------------------------------

#define ANG_F  0.34906585039886590f   // deg2rad(20)
#define MAXT_F 6.0f

// --- Kernel 1: per-batch inverse-affine params (B threads total, trivial) ---
__global__ void TCR_affine_params_kernel(const float* __restrict__ random,
                                         float* __restrict__ P,
                                         int B, int W, int H) {
    int b = blockIdx.x * blockDim.x + threadIdx.x;
    if (b >= B) return;
    float r01 = random[b];                       // same tensor drives ALL params
    float tx  = 2.0f * MAXT_F * r01 - MAXT_F;
    float ty  = tx;
    float r   = 2.0f * ANG_F  * r01 - ANG_F;     // |r| <= 0.349 rad
    float z   = 1.0f;                            // MAX_Z == MIN_Z == 1
    float hx  = r, hy = r;                       // same random => hx == hy == r
    float a   = hx - r;                          // == 0 exactly
    float bb  = hy + r;                          // == 2r, |bb| <= 0.698 rad
    // Hardware transcendentals: exact-range for these tiny arguments.
    float cos_hx = __cosf(hx), cos_hy = __cosf(hy);
    float ca = __cosf(a),  sa = __sinf(a);       // == 1, 0 (kept literal)
    float cb = __cosf(bb), sb = __sinf(bb);
    float Wf = (float)W, Hf = (float)H;

    float T11 = z * ca / cos_hx;
    float T12 = z * sa / cos_hx;
    float T13 = (Wf * cos_hx - Wf * z * ca + 2.0f * tx * z * ca
                 - Hf * z * sa + 2.0f * ty * z * sa) / (2.0f * cos_hx);
    float T21 = z * sb / cos_hy;
    float T22 = z * cb / cos_hy;
    float T23 = (Hf * cos_hy - Wf * z * cb + 2.0f * ty * z * cb
                 - Wf * z * sb + 2.0f * tx * z * sb) / (2.0f * cos_hy);

    float det = T11 * T22 - T12 * T21;           // invert 2x2 (sampling is M^-1 @ dst)
    float* p = P + b * 8;                        // 8-float stride: aligned
    p[0] =  T22 / det;                           // i00
    p[1] = -T12 / det;                           // i01
    p[2] = -T21 / det;                           // i10
    p[3] =  T11 / det;                           // i11
    p[4] =  T13;
    p[5] =  T23;
}

// --- Kernel 2: one block per output row; 12 gathers + 3 NT stores / thread ---
__global__ __launch_bounds__(256)
void TCR_warp_affine_kernel(const float* __restrict__ img,
                            const float* __restrict__ P,
                            float* __restrict__ out,
                            int W, int H, int logH) {
    const int x = threadIdx.x;
    const int y = blockIdx.x & (H - 1);          // H is a power of two
    const int b = blockIdx.x >> logH;

    // b is wave-uniform (pure SALU from ttmp9) -> scalar param reads.
    const float* p = P + b * 8;
    const float i00 = p[0], i01 = p[1], i10 = p[2], i11 = p[3];
    const float T13 = p[4], T23 = p[5];

    // Source coordinates (align_corners pixel space).
    const float dx = (float)x - T13;
    const float dy = (float)y - T23;
    const float sx = i00 * dx + i01 * dy;
    const float sy = i10 * dx + i11 * dy;

    const float x0f = floorf(sx), y0f = floorf(sy);
    const float wx = sx - x0f,    wy = sy - y0f;
    const int x0 = (int)x0f, y0 = (int)y0f;
    const int x1 = x0 + 1,   y1 = y0 + 1;

    // Validity (unsigned compare folds the >=0 and <N checks).
    const bool vx0 = (unsigned)x0 < (unsigned)W;
    const bool vx1 = (unsigned)x1 < (unsigned)W;
    const bool vy0 = (unsigned)y0 < (unsigned)H;
    const bool vy1 = (unsigned)y1 < (unsigned)H;
    const bool ok00 = vx0 && vy0, ok01 = vx1 && vy0;
    const bool ok10 = vx0 && vy1, ok11 = vx1 && vy1;

    // Clipped (always in-bounds) gather addresses -> loads are unconditional,
    // zeros applied with v_cndmask (no exec-mask branches on the border).
    const int xc0 = min(max(x0, 0), W - 1), xc1 = min(max(x1, 0), W - 1);
    const int yc0 = min(max(y0, 0), H - 1), yc1 = min(max(y1, 0), H - 1);
    const int i00x = yc0 * W + xc0, i01x = yc0 * W + xc1;
    const int i10x = yc1 * W + xc0, i11x = yc1 * W + xc1;

    const float w00 = (1.0f - wx) * (1.0f - wy);
    const float w01 = wx * (1.0f - wy);
    const float w10 = (1.0f - wx) * wy;
    const float w11 = wx * wy;

    const int plane = W * H;
    const float* pc = img + (size_t)(b * 3) * plane;
    float*       po = out + (size_t)(b * 3) * plane + y * W + x;

    // Speculative prefetch of channels 1 and 2 (clipped addrs: always valid).
    __builtin_prefetch(pc + plane     + i00x, 0, 1);   // -> global_prefetch_b8
    __builtin_prefetch(pc + 2 * plane + i00x, 0, 1);

#pragma unroll
    for (int c = 0; c < 3; ++c) {
        float v00 = pc[i00x]; v00 = ok00 ? v00 : 0.0f;
        float v01 = pc[i01x]; v01 = ok01 ? v01 : 0.0f;
        float v10 = pc[i10x]; v10 = ok10 ? v10 : 0.0f;
        float v11 = pc[i11x]; v11 = ok11 ? v11 : 0.0f;
        float res = fmaf(v00, w00, fmaf(v01, w01, fmaf(v10, w10, v11 * w11)));
        // Output is never re-read: non-temporal store (TH=NT) keeps the 192MB
        // L2 free to hold the entire 100MB input image for the gathers.
        __builtin_nontemporal_store(res, po);
        pc += plane;
        po += plane;
    }
}

extern "C" void kernel_launch(void* const* d_in, const int* in_sizes, int n_in,
                              void* d_out, int out_size, void* d_ws, size_t ws_size,
                              hipStream_t stream) {
    const float* img = (const float*)d_in[0];   // [B, 3, H, W] fp32
    const float* rnd = (const float*)d_in[1];   // [B] fp32
    float*       out = (float*)d_out;

    const int B = in_sizes[1];                  // 128
    const int C = 3;
    const int plane = in_sizes[0] / (B * C);    // H*W (square, power-of-two dims)
    int W = 1, logW = 0;
    while (W * W < plane) { W <<= 1; ++logW; }  // 256, 8
    const int H = plane >> logW;                // 256
    int logH = 0;
    while ((1 << logH) < H) ++logH;             // 8

    float* P = (float*)d_ws;                    // B * 8 floats of scratch

    TCR_affine_params_kernel<<<(B + 127) / 128, 128, 0, stream>>>(rnd, P, B, W, H);
    TCR_warp_affine_kernel<<<B * H, W, 0, stream>>>(img, P, out, W, H, logH);
}